// ASTRAContrastiveLoss_44770739094063
// MI455X (gfx1250) — compile-verified
//
#include <hip/hip_runtime.h>
#include <hip/hip_bf16.h>

typedef __attribute__((ext_vector_type(2))) float v2f;
typedef __attribute__((ext_vector_type(4))) float v4f;
typedef __attribute__((ext_vector_type(8))) float v8f;

#define D_DIM          256
#define N_AGENTS       64
#define B_SAMPLES      1024
#define ROWS_TOTAL     (B_SAMPLES * N_AGENTS)   /* 65536 */
#define ROWS_PER_WAVE  16
#define WAVES_PER_BLK  8
#define ROWS_PER_BLK   (ROWS_PER_WAVE * WAVES_PER_BLK) /* 128 */

// ---------------------------------------------------------------------------
// Kernel 1: per-(b,n) cosine similarity via WMMA K-reduction (ones-matrix trick)
//   D[i][j] += sum_k A[i][k] * 1   -> per-row partial sums live in the C operand
//
// Lane L feeds row (L & 15) for both lane halves (ISA A-layout). Because the
// B operand is all-ones, the result only depends on the SET of elements each
// row contributes across the loop, not on which K-slot they occupy. So each
// lane streams a contiguous float4 per iteration (global_load_b128):
//   lanes 0-15  cover elements [8c .. 8c+3]  of their row,
//   lanes 16-31 cover elements [8c+4 .. 8c+7],
// consumed as two v2f A-operands in two back-to-back WMMAs. Every element of
// the row is summed exactly once -> exact fp32 dot (reassociated order).
// ---------------------------------------------------------------------------
__global__ __launch_bounds__(256) void astra_cos_wmma(
    const float* __restrict__ emb_mut,
    const float* __restrict__ emb_heal,
    float* __restrict__ cos_out)
{
    const int lane    = threadIdx.x & 31;
    const int wave    = threadIdx.x >> 5;
    const int rowBase = blockIdx.x * ROWS_PER_BLK + wave * ROWS_PER_WAVE;
    const int r       = lane & 15;   // matrix row this lane feeds
    const int khalf   = lane >> 4;   // which contiguous half of the 8-elem slab

    const size_t rowOff = (size_t)(rowBase + r) * D_DIM + (size_t)(khalf * 4);
    const float* pm = emb_mut  + rowOff;
    const float* ph = emb_heal + rowOff;

    v8f c_dot = {};  // accumulates mut . heal  per row
    v8f c_m2  = {};  // accumulates |mut|^2     per row
    v8f c_h2  = {};  // accumulates |heal|^2    per row

    v2f ones; ones[0] = 1.0f; ones[1] = 1.0f;

    #pragma unroll 2
    for (int c = 0; c < D_DIM / 8; ++c) {
        v4f m = *(const v4f*)(pm + c * 8);   // global_load_b128
        v4f h = *(const v4f*)(ph + c * 8);   // global_load_b128

        v4f mh = m * h;
        v4f mm = m * m;
        v4f hh = h * h;

        v2f mh0; mh0[0] = mh[0]; mh0[1] = mh[1];
        v2f mm0; mm0[0] = mm[0]; mm0[1] = mm[1];
        v2f hh0; hh0[0] = hh[0]; hh0[1] = hh[1];
        v2f mh1; mh1[0] = mh[2]; mh1[1] = mh[3];
        v2f mm1; mm1[0] = mm[2]; mm1[1] = mm[3];
        v2f hh1; hh1[0] = hh[2]; hh1[1] = hh[3];

        // 8 args: (neg_a, A, neg_b, B, c_mod, C, reuse_a, reuse_b)
        c_dot = __builtin_amdgcn_wmma_f32_16x16x4_f32(false, mh0, false, ones,
                                                      (short)0, c_dot, false, false);
        c_m2  = __builtin_amdgcn_wmma_f32_16x16x4_f32(false, mm0, false, ones,
                                                      (short)0, c_m2,  false, false);
        c_h2  = __builtin_amdgcn_wmma_f32_16x16x4_f32(false, hh0, false, ones,
                                                      (short)0, c_h2,  false, false);
        c_dot = __builtin_amdgcn_wmma_f32_16x16x4_f32(false, mh1, false, ones,
                                                      (short)0, c_dot, false, false);
        c_m2  = __builtin_amdgcn_wmma_f32_16x16x4_f32(false, mm1, false, ones,
                                                      (short)0, c_m2,  false, false);
        c_h2  = __builtin_amdgcn_wmma_f32_16x16x4_f32(false, hh1, false, ones,
                                                      (short)0, c_h2,  false, false);
    }

    // C/D layout: VGPR p, lanes 0-15 -> M=p ; lanes 16-31 -> M=p+8 (N = any,
    // since B was all-ones every column is identical). Lane 0 emits rows 0-7,
    // lane 16 emits rows 8-15.
    if (r == 0) {
        const int mbase = khalf * 8;
        #pragma unroll
        for (int p = 0; p < 8; ++p) {
            float d  = c_dot[p];
            float nm = fmaxf(sqrtf(c_m2[p]), 1e-8f);
            float nh = fmaxf(sqrtf(c_h2[p]), 1e-8f);
            cos_out[rowBase + mbase + p] = d / (nm * nh);
        }
    }
}

// ---------------------------------------------------------------------------
// Kernel 2: per-sample loss + deterministic single-block tree reduction.
// One thread per sample (1024 threads = 32 waves on one WGP).
// ---------------------------------------------------------------------------
__global__ __launch_bounds__(1024) void astra_loss_reduce(
    const float* __restrict__ cosBuf,
    const int* __restrict__ mistake_idx,
    const unsigned char* __restrict__ agent_mask,
    float* __restrict__ out)
{
    __shared__ float sTot[B_SAMPLES];
    __shared__ float sCnt[B_SAMPLES];

    const int b  = threadIdx.x;
    const int id = mistake_idx[b];
    const bool valid = (id >= 0) && (id < N_AGENTS);
    const int  ic    = id < 0 ? 0 : (id > N_AGENTS - 1 ? N_AGENTS - 1 : id);

    const float*         row  = cosBuf     + (size_t)b * N_AGENTS;
    const unsigned char* mrow = agent_mask + (size_t)b * N_AGENTS;

    float sum = 0.0f;
    float cnt = 0.0f;
    #pragma unroll
    for (int n = 0; n < N_AGENTS; ++n) {
        float c  = row[n];
        bool  nm = (mrow[n] != 0) && (n != ic);
        sum += nm ? (1.0f - c) : 0.0f;
        cnt += nm ? 1.0f : 0.0f;
    }

    float cos_t = row[ic];
    float loss_t = fmaxf(cos_t + 1.0f, 0.0f);              // relu(cos + MARGIN)
    float loss_o = (cnt > 0.0f) ? (sum / cnt) : 0.0f;
    float per    = 0.7f * loss_t + 0.3f * loss_o;          // ALPHA blend

    sTot[b] = valid ? per  : 0.0f;
    sCnt[b] = valid ? 1.0f : 0.0f;
    __syncthreads();

    #pragma unroll
    for (int s = B_SAMPLES / 2; s > 0; s >>= 1) {
        if (b < s) {
            sTot[b] += sTot[b + s];
            sCnt[b] += sCnt[b + s];
        }
        __syncthreads();
    }

    if (b == 0) {
        float vc = sCnt[0];
        out[0] = (vc > 0.0f) ? (sTot[0] / vc) : 0.0f;
    }
}

// ---------------------------------------------------------------------------
extern "C" void kernel_launch(void* const* d_in, const int* in_sizes, int n_in,
                              void* d_out, int out_size, void* d_ws, size_t ws_size,
                              hipStream_t stream)
{
    const float*         emb_mut  = (const float*)d_in[0];
    const float*         emb_heal = (const float*)d_in[1];
    const int*           idx      = (const int*)d_in[2];
    const unsigned char* mask     = (const unsigned char*)d_in[3];
    float*               out      = (float*)d_out;

    // workspace: cos[B*N] floats = 256 KB
    float* cosBuf = (float*)d_ws;

    astra_cos_wmma<<<ROWS_TOTAL / ROWS_PER_BLK, 256, 0, stream>>>(
        emb_mut, emb_heal, cosBuf);
    astra_loss_reduce<<<1, B_SAMPLES, 0, stream>>>(cosBuf, idx, mask, out);

    (void)in_sizes; (void)n_in; (void)out_size; (void)ws_size;
}